// Sampler_34900904247420
// MI455X (gfx1250) — compile-verified
//
#include <hip/hip_runtime.h>
#include <hip/hip_bf16.h>
#include <stdint.h>

// ---------------------------------------------------------------------------
// Top-k (k=50) Gumbel-max sampler, B=256 rows x V=128000 vocab, fp32.
// Bandwidth bound: 131MB input; pass 1 streams from HBM via the gfx1250
// async->LDS pipeline (global_load_async_to_lds_b128, ASYNCcnt), 3 tiles in
// flight to hide HBM latency. Refinement + sampling passes re-read from the
// 192MB L2 (entire input is L2-resident). No softmax pass needed:
// argmax(probs/e) == argmax_{kept}(scaled - log e)  (Gumbel-max).
// ---------------------------------------------------------------------------

#ifndef __has_builtin
#define __has_builtin(x) 0
#endif

#if __has_builtin(__builtin_amdgcn_global_load_async_to_lds_b128)
#define HAVE_ASYNC_LDS 1
#else
#define HAVE_ASYNC_LDS 0
#endif

#if __has_builtin(__builtin_amdgcn_s_wait_asynccnt)
#define HAVE_WAIT_ASYNC 1
#else
#define HAVE_WAIT_ASYNC 0
#endif

#define BLOCK 1024           // 32 wave32 per workgroup, 1 workgroup per row
#define PIPE_DEPTH 3         // async tiles in flight (buffers = PIPE_DEPTH+1)
#define GLOBAL_AS __attribute__((address_space(1)))
#define LDS_AS    __attribute__((address_space(3)))

// Builtin signature (from hipcc diagnostics): (v4i AS1*, v4i AS3*, imm, imm).
typedef int v4i __attribute__((vector_size(4 * sizeof(int))));

template <int N>
__device__ inline void async_wait() {
#if HAVE_WAIT_ASYNC
    __builtin_amdgcn_s_wait_asynccnt(N);
#else
    asm volatile("s_wait_asynccnt %0" :: "i"(N));
#endif
    asm volatile("" ::: "memory");
}

// Monotonic float -> uint32 key (ascending order preserved).
__device__ inline uint32_t fkey(float f) {
    uint32_t u = __float_as_uint(f);
    return (u & 0x80000000u) ? ~u : (u | 0x80000000u);
}

__device__ inline uint32_t rotl32(uint32_t x, int r) { return (x << r) | (x >> (32 - r)); }

// Threefry-2x32, 20 rounds, key (0, 42) — matches jax.random.key(42).
__device__ inline uint32_t threefry2x32_word(uint32_t x0, uint32_t x1, bool second) {
    const uint32_t k0 = 0u, k1 = 42u;
    const uint32_t k2 = 0x1BD11BDAu ^ k0 ^ k1;
    x0 += k0; x1 += k1;
#define TFR(r) { x0 += x1; x1 = rotl32(x1, r); x1 ^= x0; }
    TFR(13) TFR(15) TFR(26) TFR(6)   x0 += k1; x1 += k2 + 1u;
    TFR(17) TFR(29) TFR(16) TFR(24)  x0 += k2; x1 += k0 + 2u;
    TFR(13) TFR(15) TFR(26) TFR(6)   x0 += k0; x1 += k1 + 3u;
    TFR(17) TFR(29) TFR(16) TFR(24)  x0 += k1; x1 += k2 + 4u;
    TFR(13) TFR(15) TFR(26) TFR(6)   x0 += k2; x1 += k0 + 5u;
#undef TFR
    return second ? x1 : x0;
}

// -log(e) where e = jax.random.exponential(key(42), (B*V,))[flat].
// JAX splits iota counts in halves: out[i<h] = word0(pair(i, i+h)),
// out[i>=h] = word1(pair(i-h, i)).
__device__ inline float gumbel_at(uint32_t flat, uint32_t half_n) {
    uint32_t bits;
    if (flat < half_n) bits = threefry2x32_word(flat, flat + half_n, false);
    else               bits = threefry2x32_word(flat - half_n, flat, true);
    float u = __uint_as_float((bits >> 9) | 0x3f800000u) - 1.0f;  // [0,1)
    float e = -log1pf(-u);                                        // Exp(1)
    return -logf(e);                                              // Gumbel
}

// In-place suffix-sum over hist[0..nbins) then pick the unique bin b with
// suffix[b] >= rem > suffix[b+1]. Returns via shared s_bin / s_above.
__device__ inline void suffix_select(uint32_t* hist, int nbins, uint32_t rem,
                                     int tid, int* s_bin, uint32_t* s_above) {
    for (int off = 1; off < nbins; off <<= 1) {
        uint32_t a0 = 0u, a1 = 0u;
        int i0 = tid, i1 = tid + BLOCK;
        if (i0 + off < nbins) a0 = hist[i0 + off];
        if (i1 < nbins && i1 + off < nbins) a1 = hist[i1 + off];
        __syncthreads();
        if (i0 < nbins) hist[i0] += a0;
        if (i1 < nbins) hist[i1] += a1;
        __syncthreads();
    }
    for (int b = tid; b < nbins; b += BLOCK) {
        uint32_t sb = hist[b];
        uint32_t sn = (b + 1 < nbins) ? hist[b + 1] : 0u;
        if (sb >= rem && sn < rem) { *s_bin = b; *s_above = sn; }
    }
    __syncthreads();
}

__device__ inline void pass1_elem(float v, int idx, uint32_t* hist,
                                  unsigned long long& best) {
    uint32_t key = fkey(v);
    atomicAdd(&hist[key >> 21], 1u);
    unsigned long long p =
        ((unsigned long long)key << 32) | (uint32_t)~(uint32_t)idx;  // ~idx: ties -> smallest idx
    if (p > best) best = p;
}

__global__ __launch_bounds__(BLOCK)
void Sampler_34900904247420_kernel(const float* __restrict__ logits,
                                   const float* __restrict__ temps,
                                   const int* __restrict__ topk_ptr,
                                   int* __restrict__ out, int B, int V) {
    const int row = blockIdx.x;
    const int tid = threadIdx.x;
    const float* rp = logits + (size_t)row * (size_t)V;
    const float4* rp4 = reinterpret_cast<const float4*>(rp);

    __shared__ uint32_t hist[2048];
    __shared__ unsigned long long s_greedy;
    __shared__ unsigned long long s_sample;
    __shared__ int s_bin;
    __shared__ uint32_t s_above;
#if HAVE_ASYNC_LDS
    __shared__ __align__(16) float tile[PIPE_DEPTH + 1][4 * BLOCK];  // 4 x 16KB
#endif

    if (tid == 0) { s_greedy = 0ull; s_sample = 0ull; }
    for (int b = tid; b < 2048; b += BLOCK) hist[b] = 0u;
    __syncthreads();

    // ---- Pass 1 (HBM): level-1 histogram (key[31:21]) + argmax ------------
    unsigned long long best = 0ull;
#if HAVE_ASYNC_LDS
    {
        const int TILE = 4 * BLOCK;                     // floats per tile
        const int NT = (V + TILE - 1) / TILE;
        const int NBUF = PIPE_DEPTH + 1;
        // prime PIPE_DEPTH tiles
        for (int p = 0; p < PIPE_DEPTH && p < NT; ++p) {
            int gi = p * TILE + tid * 4; if (gi > V - 4) gi = V - 4;
            __builtin_amdgcn_global_load_async_to_lds_b128(
                (GLOBAL_AS v4i*)(rp + gi),
                (LDS_AS v4i*)(&tile[p % NBUF][tid * 4]), 0, 0);
        }
        for (int t = 0; t < NT; ++t) {
            int cur = t % NBUF;
            if (t + PIPE_DEPTH < NT) {
                int gi = (t + PIPE_DEPTH) * TILE + tid * 4; if (gi > V - 4) gi = V - 4;
                __builtin_amdgcn_global_load_async_to_lds_b128(
                    (GLOBAL_AS v4i*)(rp + gi),
                    (LDS_AS v4i*)(&tile[(t + PIPE_DEPTH) % NBUF][tid * 4]), 0, 0);
                async_wait<PIPE_DEPTH>();   // tile t complete, PIPE_DEPTH in flight
            } else {
                async_wait<0>();            // tail: drain (last few tiles only)
            }
            int gi = t * TILE + tid * 4;
            if (gi < V) {                   // each lane reads its own slot
                float4 v = *reinterpret_cast<const float4*>(&tile[cur][tid * 4]);
                pass1_elem(v.x, gi + 0, hist, best);
                pass1_elem(v.y, gi + 1, hist, best);
                pass1_elem(v.z, gi + 2, hist, best);
                pass1_elem(v.w, gi + 3, hist, best);
            }
        }
    }
#else
    for (int i = tid * 4; i < V; i += 4 * BLOCK) {
        float4 v = rp4[i >> 2];
        pass1_elem(v.x, i + 0, hist, best);
        pass1_elem(v.y, i + 1, hist, best);
        pass1_elem(v.z, i + 2, hist, best);
        pass1_elem(v.w, i + 3, hist, best);
    }
#endif
    atomicMax(&s_greedy, best);
    __syncthreads();

    // ---- Radix selection of exact K-th largest key ------------------------
    int K = topk_ptr ? *topk_ptr : 50;
    const bool no_topk = (K <= 0);
    if (K > V) K = V;
    uint32_t thrKey = 0u;                               // keep-all if no top-k
    if (!no_topk) {                                     // uniform branch
        uint32_t rem = (uint32_t)K;
        suffix_select(hist, 2048, rem, tid, &s_bin, &s_above);
        const uint32_t b1 = (uint32_t)s_bin; rem -= s_above;

        for (int b = tid; b < 2048; b += BLOCK) hist[b] = 0u;
        __syncthreads();
        // ---- Pass 2 (L2): refine key[20:10] within bin b1 -----------------
        for (int i = tid * 4; i < V; i += 4 * BLOCK) {
            if (i + 4 * BLOCK < V) __builtin_prefetch(rp + i + 4 * BLOCK, 0, 0);
            float4 v = rp4[i >> 2];
            uint32_t k0 = fkey(v.x), k1 = fkey(v.y), k2 = fkey(v.z), k3 = fkey(v.w);
            if ((k0 >> 21) == b1) atomicAdd(&hist[(k0 >> 10) & 2047u], 1u);
            if ((k1 >> 21) == b1) atomicAdd(&hist[(k1 >> 10) & 2047u], 1u);
            if ((k2 >> 21) == b1) atomicAdd(&hist[(k2 >> 10) & 2047u], 1u);
            if ((k3 >> 21) == b1) atomicAdd(&hist[(k3 >> 10) & 2047u], 1u);
        }
        __syncthreads();
        suffix_select(hist, 2048, rem, tid, &s_bin, &s_above);
        const uint32_t b2 = (uint32_t)s_bin; rem -= s_above;
        const uint32_t pref22 = (b1 << 11) | b2;        // key[31:10]

        for (int b = tid; b < 1024; b += BLOCK) hist[b] = 0u;
        __syncthreads();
        // ---- Pass 3 (L2): refine key[9:0] ---------------------------------
        for (int i = tid * 4; i < V; i += 4 * BLOCK) {
            float4 v = rp4[i >> 2];
            uint32_t k0 = fkey(v.x), k1 = fkey(v.y), k2 = fkey(v.z), k3 = fkey(v.w);
            if ((k0 >> 10) == pref22) atomicAdd(&hist[k0 & 1023u], 1u);
            if ((k1 >> 10) == pref22) atomicAdd(&hist[k1 & 1023u], 1u);
            if ((k2 >> 10) == pref22) atomicAdd(&hist[k2 & 1023u], 1u);
            if ((k3 >> 10) == pref22) atomicAdd(&hist[k3 & 1023u], 1u);
        }
        __syncthreads();
        suffix_select(hist, 1024, rem, tid, &s_bin, &s_above);
        thrKey = (pref22 << 10) | (uint32_t)s_bin;      // exact K-th largest key
    }
    __syncthreads();

    // ---- Pass 4 (L2): Gumbel-max over kept elements -----------------------
    const float t = temps[row];
    const float safe_t = (t == 0.0f) ? 1.0f : t;
    const uint32_t half_n = (uint32_t)(((unsigned long long)B * (unsigned long long)V) >> 1);
    unsigned long long sbest = 0ull;
    for (int i = tid * 4; i < V; i += 4 * BLOCK) {
        if (i + 4 * BLOCK < V) __builtin_prefetch(rp + i + 4 * BLOCK, 0, 0);
        float4 v = rp4[i >> 2];
        const float vv[4] = {v.x, v.y, v.z, v.w};
#pragma unroll
        for (int j = 0; j < 4; ++j) {
            uint32_t key = fkey(vv[j]);
            if (key >= thrKey) {                        // ~K elements per row reach here
                float scaled = vv[j] / safe_t;
                float y = scaled + gumbel_at((uint32_t)(row * V + i + j), half_n);
                unsigned long long p =
                    ((unsigned long long)fkey(y) << 32) | (uint32_t)~(uint32_t)(i + j);
                if (p > sbest) sbest = p;
            }
        }
    }
    if (sbest) atomicMax(&s_sample, sbest);
    __syncthreads();

    if (tid == 0) {
        uint32_t gidx = ~(uint32_t)(s_greedy & 0xffffffffull);
        uint32_t sidx = ~(uint32_t)(s_sample & 0xffffffffull);
        out[row] = (t == 0.0f) ? (int)gidx : (int)sidx;
    }
}

extern "C" void kernel_launch(void* const* d_in, const int* in_sizes, int n_in,
                              void* d_out, int out_size, void* d_ws, size_t ws_size,
                              hipStream_t stream) {
    const float* logits = (const float*)d_in[0];
    const float* temps  = (const float*)d_in[1];
    const int*   topk   = (n_in > 2) ? (const int*)d_in[2] : nullptr;
    const int B = in_sizes[1];
    const int V = in_sizes[0] / B;
    Sampler_34900904247420_kernel<<<dim3(B), dim3(BLOCK), 0, stream>>>(
        logits, temps, topk, (int*)d_out, B, V);
}